// MoERouter_44409961840862
// MI455X (gfx1250) — compile-verified
//
#include <hip/hip_runtime.h>
#include <hip/hip_bf16.h>
#include <math.h>

typedef __attribute__((ext_vector_type(2))) float v2f;
typedef __attribute__((ext_vector_type(8))) float v8f;

#define D_MODEL 2048
#define N_EXP   64
#define KCHUNK  128          // K tile staged in LDS per buffer
#define NCHUNK  (D_MODEL / KCHUNK)
#define LDS_STRIDE 132       // 128 + 4 pad floats -> bank-conflict-free B reads
#define GEMM_WAVES 4         // 128 threads/block -> 64 tokens/block -> 256 blocks
#define TOPK_WAVES 8

// ---------------------------------------------------------------------------
// Kernel 1: logits[token, e] = x[token,:] . W[e,:]  via V_WMMA_F32_16X16X4_F32
// Each wave computes a 16(token) x 64(expert) tile: 4 accumulators of v8f.
// W K-chunks double-buffered in LDS so staging overlaps the WMMA stream.
// ---------------------------------------------------------------------------
__global__ __launch_bounds__(128) void moe_router_gemm(
    const float* __restrict__ x,      // [TOKENS, 2048]
    const float* __restrict__ W,      // [64, 2048]
    float* __restrict__ logits)       // [TOKENS, 64]
{
    __shared__ __align__(16) float wlds[2][N_EXP * LDS_STRIDE];  // 2 x 33 KB

    const int lane = threadIdx.x & 31;
    const int wave = threadIdx.x >> 5;
    const int m    = lane & 15;            // token-in-tile / expert-in-tile
    const int koff = (lane >> 4) << 1;     // 0 or 2 (K sub-offset per A/B layout)

    const int rowbase = (blockIdx.x * GEMM_WAVES + wave) * 16;
    const float* xrow = x + (size_t)(rowbase + m) * D_MODEL;

    v8f acc0 = {}, acc1 = {}, acc2 = {}, acc3 = {};

    // --- prologue: stage chunk 0 into buffer 0 (64 rows x 128 floats) ---
    #pragma unroll
    for (int i = 0; i < 16; ++i) {
        int idx = threadIdx.x + 128 * i;            // 0..2047 float4 slots
        int e   = idx >> 5;                         // 32 float4 per expert row
        int q   = idx & 31;
        *(float4*)(&wlds[0][e * LDS_STRIDE + q * 4]) =
            *(const float4*)(W + (size_t)e * D_MODEL + q * 4);
    }

    for (int chunk = 0; chunk < NCHUNK; ++chunk) {
        const int buf   = chunk & 1;
        const int kbase = chunk * KCHUNK;

        // one barrier per iteration: guarantees buf is fully staged and
        // buf^1's previous readers are done before we overwrite it below.
        __syncthreads();

        // --- stage next chunk into the other buffer (overlaps WMMAs) ---
        if (chunk + 1 < NCHUNK) {
            const int knext = kbase + KCHUNK;
            #pragma unroll
            for (int i = 0; i < 16; ++i) {
                int idx = threadIdx.x + 128 * i;
                int e   = idx >> 5;
                int q   = idx & 31;
                *(float4*)(&wlds[buf ^ 1][e * LDS_STRIDE + q * 4]) =
                    *(const float4*)(W + (size_t)e * D_MODEL + knext + q * 4);
            }
            // pull the next x chunk toward the caches (A is the HBM stream)
            __builtin_prefetch(xrow + knext, 0, 0);
        }

        // --- 32 k-steps of WMMA f32 16x16x4 over 4 expert tiles ---
        const float* wbuf = wlds[buf];
        #pragma unroll 4
        for (int s = 0; s < KCHUNK / 4; ++s) {
            const int kk = 4 * s + koff;
            // A fragment: 16x4 f32 tile of x (lane m = token row, koff = K pair)
            v2f a = *(const v2f*)(xrow + kbase + kk);
            // B fragments: 4x16 tiles of W^T == rows of W (lane m = expert col)
            v2f b0 = *(const v2f*)(wbuf + ( 0 + m) * LDS_STRIDE + kk);
            v2f b1 = *(const v2f*)(wbuf + (16 + m) * LDS_STRIDE + kk);
            v2f b2 = *(const v2f*)(wbuf + (32 + m) * LDS_STRIDE + kk);
            v2f b3 = *(const v2f*)(wbuf + (48 + m) * LDS_STRIDE + kk);
            acc0 = __builtin_amdgcn_wmma_f32_16x16x4_f32(false, a, false, b0, (short)0, acc0, false, false);
            acc1 = __builtin_amdgcn_wmma_f32_16x16x4_f32(false, a, false, b1, (short)0, acc1, false, false);
            acc2 = __builtin_amdgcn_wmma_f32_16x16x4_f32(false, a, false, b2, (short)0, acc2, false, false);
            acc3 = __builtin_amdgcn_wmma_f32_16x16x4_f32(false, a, false, b3, (short)0, acc3, false, false);
        }
    }

    // C/D layout: VGPR r, lanes 0-15 -> M=r, N=lane; lanes 16-31 -> M=r+8, N=lane-16
    const int rsel = (lane >> 4) << 3;  // +8 rows for upper half-wave
    #pragma unroll
    for (int r = 0; r < 8; ++r) {
        const size_t row = (size_t)(rowbase + r + rsel) * N_EXP;
        logits[row +  0 + m] = acc0[r];
        logits[row + 16 + m] = acc1[r];
        logits[row + 32 + m] = acc2[r];
        logits[row + 48 + m] = acc3[r];
    }
}

// ---------------------------------------------------------------------------
// Kernel 2: per-token top-2 + normalized gates + expert count accumulation.
// One wave32 per token; lane l owns experts l and l+32.
// softmax->topk->renorm collapses to: g1 = 1/(1+exp(l2-l1)), g2 = 1-g1.
// ---------------------------------------------------------------------------
__global__ __launch_bounds__(256) void moe_router_topk(
    const float* __restrict__ logits,  // [TOKENS, 64]
    float* __restrict__ gates,         // [TOKENS, 64]
    float* __restrict__ idx_out,       // [TOKENS, 2] (indices, stored as float)
    float* __restrict__ counts)        // [64]
{
    const int token = blockIdx.x * TOPK_WAVES + (threadIdx.x >> 5);
    const int lane  = threadIdx.x & 31;
    const float* lrow = logits + (size_t)token * N_EXP;

    const float v0 = lrow[lane];
    const float v1 = lrow[lane + 32];

    // ---- top-1 (argmax, tie -> lower index, matches jax.lax.top_k) ----
    float bv; int bi;
    if (v0 >= v1) { bv = v0; bi = lane; } else { bv = v1; bi = lane + 32; }
    #pragma unroll
    for (int mshift = 16; mshift >= 1; mshift >>= 1) {
        float ov = __shfl_xor(bv, mshift, 32);
        int   oi = __shfl_xor(bi, mshift, 32);
        if (ov > bv || (ov == bv && oi < bi)) { bv = ov; bi = oi; }
    }
    const int i1 = bi; const float l1 = bv;

    // ---- top-2: mask out i1, argmax again ----
    const float w0 = (lane      == i1) ? -INFINITY : v0;
    const float w1 = (lane + 32 == i1) ? -INFINITY : v1;
    if (w0 >= w1) { bv = w0; bi = lane; } else { bv = w1; bi = lane + 32; }
    #pragma unroll
    for (int mshift = 16; mshift >= 1; mshift >>= 1) {
        float ov = __shfl_xor(bv, mshift, 32);
        int   oi = __shfl_xor(bi, mshift, 32);
        if (ov > bv || (ov == bv && oi < bi)) { bv = ov; bi = oi; }
    }
    const int i2 = bi; const float l2 = bv;

    const float e  = __expf(l2 - l1);     // <= 1
    const float g2 = e / (1.0f + e);
    const float g1 = 1.0f - g2;

    const float o0 = (lane        == i1) ? g1 : ((lane        == i2) ? g2 : 0.0f);
    const float o1 = ((lane + 32) == i1) ? g1 : (((lane + 32) == i2) ? g2 : 0.0f);

    gates[(size_t)token * N_EXP + lane]      = o0;
    gates[(size_t)token * N_EXP + lane + 32] = o1;
    if (lane == 0) {
        idx_out[(size_t)token * 2 + 0] = (float)i1;
        idx_out[(size_t)token * 2 + 1] = (float)i2;
    }
    // only 2 nonzero gate values per token -> 2 atomics/token over 64 counters
    if (o0 != 0.0f) atomicAdd(counts + lane,      o0);
    if (o1 != 0.0f) atomicAdd(counts + lane + 32, o1);
}

// ---------------------------------------------------------------------------
// Kernel 0: zero the expert-count accumulator (harness does not re-poison ws)
// ---------------------------------------------------------------------------
__global__ void moe_zero_counts(float* counts) {
    counts[threadIdx.x] = 0.0f;
}

// ---------------------------------------------------------------------------
// Kernel 3: load_loss = mean((counts/total * 64 - 1)^2)
// ---------------------------------------------------------------------------
__global__ void moe_load_loss(const float* __restrict__ counts,
                              float* __restrict__ loss_out) {
    __shared__ float s[N_EXP];
    const int t = threadIdx.x;
    s[t] = counts[t];
    __syncthreads();
    if (t == 0) {
        float total = 0.0f;
        for (int i = 0; i < N_EXP; ++i) total += s[i];
        const float inv = (float)N_EXP / total;
        float loss = 0.0f;
        for (int i = 0; i < N_EXP; ++i) {
            const float d = s[i] * inv - 1.0f;
            loss += d * d;
        }
        loss_out[0] = loss / (float)N_EXP;
    }
}

extern "C" void kernel_launch(void* const* d_in, const int* in_sizes, int n_in,
                              void* d_out, int out_size, void* d_ws, size_t ws_size,
                              hipStream_t stream) {
    const float* x = (const float*)d_in[0];            // [4,4096,2048] fp32
    const float* W = (const float*)d_in[1];            // [64,2048] fp32
    // d_in[2]=n_active(=2), d_in[3]=capacity — fixed by kernel structure.

    const int TOKENS = in_sizes[0] / D_MODEL;          // 16384

    float* gates   = (float*)d_out;                    // [TOKENS,64]
    float* idx_out = gates + (size_t)TOKENS * N_EXP;   // [TOKENS,2]
    float* loss    = idx_out + (size_t)TOKENS * 2;     // [1]

    float* logits  = (float*)d_ws;                     // [TOKENS,64] = 4 MB
    float* counts  = logits + (size_t)TOKENS * N_EXP;  // [64]

    moe_zero_counts<<<1, N_EXP, 0, stream>>>(counts);

    // 4 waves/block, each wave = 16 tokens -> 64 tokens/block -> 256 blocks
    moe_router_gemm<<<TOKENS / (GEMM_WAVES * 16), 128, 0, stream>>>(x, W, logits);

    // 1 wave per token, 8 tokens/block
    moe_router_topk<<<TOKENS / TOPK_WAVES, 256, 0, stream>>>(logits, gates, idx_out, counts);

    moe_load_loss<<<1, N_EXP, 0, stream>>>(counts, loss);
}